// CCNeuron_88252987998517
// MI455X (gfx1250) — compile-verified
//
#include <hip/hip_runtime.h>

// CCNeuron batched scan, MI455X (gfx1250).
//
// Roofline: ~0.5 GFLOP vs ~0.70 GB streamed => memory-bound, ~30us floor at
// 23.3 TB/s. No GEMM structure (each of the 32768 neurons has private 2x2
// weights) => WMMA cannot help. Parallelism is capped at B=32768 threads
// (T-scan is serial) = 1024 waves ~= 1 wave/SIMD on MI455X, so latency hiding
// must come from per-wave pipelining, not occupancy:
//   - K=16-deep register pipeline: 16 steps of inputs resident in VGPRs,
//     each demand load has ~16 iterations of slack; in-flight bytes/wave
//     (~14KB) * 1024 waves ~= the 23.3TB/s x ~0.5us BW-latency product.
//   - coalesced float2/float4 LD/ST, non-temporal (TH=NT) for the
//     single-use stream, gfx1250 scale_offset addressing.
//   - global_prefetch_b8 one block (16 steps) ahead of the in-flight loads,
//     issued under ONE uniform branch per block (no per-step SALU guards).

typedef float v2f __attribute__((ext_vector_type(2)));
typedef float v4f __attribute__((ext_vector_type(4)));

#define PV_A    0.25f
#define PYC_A   0.10f
#define ADAPT_A 0.02f   // 0.1 * 0.2

constexpr int K = 16;   // software-pipeline depth (register-resident steps)

__global__ __launch_bounds__(128)
void ccneuron_scan(const float* __restrict__ x,        // [T,B,2]
                   const float* __restrict__ c,        // [T,B,2]
                   const float* __restrict__ noise_p,  // [T,B,2]
                   const float* __restrict__ noise_y,  // [T,B]
                   const float* __restrict__ w_ff,     // [B,2]
                   const float* __restrict__ w_fb,     // [B,2]
                   const float* __restrict__ w_lat,    // [B,2]
                   const float* __restrict__ w_pv_lat, // [B,2]
                   const float* __restrict__ W_pv,     // [B,2,2]
                   const float* __restrict__ rc,       // [2]
                   const float* __restrict__ pv0,      // [B,2]
                   const float* __restrict__ y0,       // [B]
                   const float* __restrict__ a0,       // [B]
                   float* __restrict__ out,            // [T,B,4]
                   int T, int B)
{
    const int b = blockIdx.x * blockDim.x + threadIdx.x;
    if (b >= B) return;

    // ---- per-neuron constants (read once, live in VGPRs) ----
    const v2f wff = *((const v2f*)w_ff + b);
    v2f wfb       = *((const v2f*)w_fb + b);
    wfb.x *= rc[0];                 // fold receives_context into w_fb
    wfb.y *= rc[1];
    const v2f wlat = *((const v2f*)w_lat + b);
    const v2f wpl  = *((const v2f*)w_pv_lat + b);
    const v4f Wpv  = *((const v4f*)W_pv + b);   // [W00, W01, W10, W11]

    // ---- carry state ----
    v2f   pv = *((const v2f*)pv0 + b);
    float y  = y0[b];
    float a  = a0[b];

    const v2f* __restrict__ xp  = (const v2f*)x;
    const v2f* __restrict__ cp  = (const v2f*)c;
    const v2f* __restrict__ npp = (const v2f*)noise_p;
    const float* __restrict__ nyp = noise_y;
    v4f* __restrict__ op = (v4f*)out;

    // one simulation step: consumes one step's inputs, updates carry, stores.
    auto step = [&](v2f xv, v2f cv, v2f nv, float nyv, int sidx) {
        // PV pathway: pv = EMA(relu(W_pv@x + y*w_pv_lat + noise_p))
        float pvff0 = fmaf(Wpv.y, xv.y, Wpv.x * xv.x);
        float pvff1 = fmaf(Wpv.w, xv.y, Wpv.z * xv.x);
        float d0 = fmaxf(pvff0 + fmaf(y, wpl.x, nv.x), 0.0f);
        float d1 = fmaxf(pvff1 + fmaf(y, wpl.y, nv.y), 0.0f);
        pv.x = fmaf(PV_A, d0, (1.0f - PV_A) * pv.x);
        pv.y = fmaf(PV_A, d1, (1.0f - PV_A) * pv.y);
        // adaptation EMA (uses previous y)
        const float a_new = fmaf(ADAPT_A, y, (1.0f - ADAPT_A) * a);
        // pyramidal: y = EMA(relu(wff.x + wfb.c - wlat.pv + ny - a_new))
        const float yff  = fmaf(wff.y, xv.y, wff.x * xv.x);
        const float yfb  = fmaf(wfb.y, cv.y, wfb.x * cv.x);
        const float ylat = fmaf(wlat.y, pv.y, wlat.x * pv.x);
        const float ydrive = fmaxf(yff + yfb - ylat + nyv - a_new, 0.0f);
        const float y_new  = fmaf(PYC_A, ydrive, (1.0f - PYC_A) * y);
        // pack (y_prev, y_next, pv_new) -> one coalesced 16B NT store
        v4f o;
        o.x = y; o.y = y_new; o.z = pv.x; o.w = pv.y;
        __builtin_nontemporal_store(o, op + sidx);
        y = y_new; a = a_new;
    };

    const int idx0 = b;    // element index of (t=0, b) for every array

    if ((T % K == 0) && (T >= 2 * K)) {
        // =========== deep-pipelined path (T=512 takes this) ===========
        v2f   xb[K], cb[K], nb[K];
        float nyb[K];

        // prologue: fill the K-slot register buffer with steps 0..K-1
        int lidx = idx0;
        #pragma unroll
        for (int j = 0; j < K; ++j) {
            xb[j]  = __builtin_nontemporal_load(xp  + lidx);
            cb[j]  = __builtin_nontemporal_load(cp  + lidx);
            nb[j]  = __builtin_nontemporal_load(npp + lidx);
            nyb[j] = __builtin_nontemporal_load(nyp + lidx);
            lidx += B;
        }

        int sidx = idx0;
        int t = 0;
        // main: compute block [t, t+K), load block [t+K, t+2K)
        for (; t + 2 * K <= T; t += K) {
            // prefetch block [t+2K, t+3K) into L2 -- ONE uniform branch/block
            if (t + 3 * K <= T) {
                int pidx = sidx + 2 * K * B;
                #pragma unroll
                for (int j = 0; j < K; ++j) {
                    __builtin_prefetch((const void*)(xp  + pidx), 0, 0);
                    __builtin_prefetch((const void*)(cp  + pidx), 0, 0);
                    __builtin_prefetch((const void*)(npp + pidx), 0, 0);
                    __builtin_prefetch((const void*)(nyp + pidx), 0, 0);
                    pidx += B;
                }
            }
            #pragma unroll
            for (int j = 0; j < K; ++j) {
                step(xb[j], cb[j], nb[j], nyb[j], sidx);
                // refill slot j with step t+K+j (always in bounds here)
                xb[j]  = __builtin_nontemporal_load(xp  + lidx);
                cb[j]  = __builtin_nontemporal_load(cp  + lidx);
                nb[j]  = __builtin_nontemporal_load(npp + lidx);
                nyb[j] = __builtin_nontemporal_load(nyp + lidx);
                sidx += B; lidx += B;
            }
        }
        // epilogue: exactly K steps remain in the buffers (T % K == 0)
        #pragma unroll
        for (int j = 0; j < K; ++j) {
            step(xb[j], cb[j], nb[j], nyb[j], sidx);
            sidx += B;
        }
    } else {
        // =========== generic fallback (any T) ===========
        int idx = idx0;
        for (int t = 0; t < T; ++t) {
            v2f   xv  = __builtin_nontemporal_load(xp  + idx);
            v2f   cv  = __builtin_nontemporal_load(cp  + idx);
            v2f   nv  = __builtin_nontemporal_load(npp + idx);
            float nyv = __builtin_nontemporal_load(nyp + idx);
            step(xv, cv, nv, nyv, idx);
            idx += B;
        }
    }
}

extern "C" void kernel_launch(void* const* d_in, const int* in_sizes, int n_in,
                              void* d_out, int out_size, void* d_ws, size_t ws_size,
                              hipStream_t stream) {
    (void)n_in; (void)d_ws; (void)ws_size; (void)out_size;
    const float* x        = (const float*)d_in[0];
    const float* c        = (const float*)d_in[1];
    const float* noise_p  = (const float*)d_in[2];
    const float* noise_y  = (const float*)d_in[3];
    const float* w_ff     = (const float*)d_in[4];
    const float* w_fb     = (const float*)d_in[5];
    const float* w_lat    = (const float*)d_in[6];
    const float* w_pv_lat = (const float*)d_in[7];
    const float* W_pv     = (const float*)d_in[8];
    const float* rc       = (const float*)d_in[9];
    const float* pv0      = (const float*)d_in[10];
    const float* y0       = (const float*)d_in[11];
    const float* a0       = (const float*)d_in[12];
    float* out = (float*)d_out;

    const int B = in_sizes[11];              // y0 is [B]
    const int T = in_sizes[3] / B;           // noise_y is [T,B]

    const int block = 128;                   // 4 waves (wave32) per workgroup
    const int grid  = (B + block - 1) / block;
    ccneuron_scan<<<grid, block, 0, stream>>>(x, c, noise_p, noise_y,
                                              w_ff, w_fb, w_lat, w_pv_lat, W_pv,
                                              rc, pv0, y0, a0, out, T, B);
}